// Model_77077483094905
// MI455X (gfx1250) — compile-verified
//
#include <hip/hip_runtime.h>

typedef float v2f __attribute__((ext_vector_type(2)));
typedef float v8f __attribute__((ext_vector_type(8)));

#define EPS_BN 1e-4f

static constexpr int BB = 512;

// =====================================================================
// Scalar helper kernels
// =====================================================================
__global__ void init_first_k(int* __restrict__ f, int n) {
    int i = blockIdx.x * blockDim.x + threadIdx.x;
    if (i < n) f[i] = 0x7fffffff;
}

__global__ void scatter_k(const int* __restrict__ coords, int* __restrict__ f, int npts) {
    int j = blockIdx.x * blockDim.x + threadIdx.x;
    if (j >= npts) return;
    int y = coords[3 * j + 0];
    int x = coords[3 * j + 1];
    int b = coords[3 * j + 2];
    atomicMin(&f[(b * 95 + y) * 95 + x], j);
}

// keep FIRST feature at duplicate coords; dense [512,95,95,4] (4th ch = 0 pad)
__global__ void build_x0_k(const int* __restrict__ f, const float* __restrict__ feats,
                           float* __restrict__ x0, unsigned char* __restrict__ m0, int n) {
    int i = blockIdx.x * blockDim.x + threadIdx.x;
    if (i >= n) return;
    int j = f[i];
    bool act = (j != 0x7fffffff);
    m0[i] = act ? 1 : 0;
    int jc = act ? j : 0;
    float a = feats[3 * jc + 0];
    float b = feats[3 * jc + 1];
    float c = feats[3 * jc + 2];
    x0[4 * i + 0] = act ? a : 0.f;
    x0[4 * i + 1] = act ? b : 0.f;
    x0[4 * i + 2] = act ? c : 0.f;
    x0[4 * i + 3] = 0.f;  // channel pad (weights padded with zeros too)
}

// sparse maxpool 3x3 s2 VALID: [512,95,95,8] -> [512,47,47,8] + new mask
__global__ void maxpool_k(const float* __restrict__ in, const unsigned char* __restrict__ m,
                          float* __restrict__ out, unsigned char* __restrict__ mo) {
    int p = blockIdx.x * blockDim.x + threadIdx.x;
    if (p >= BB * 47 * 47) return;
    int b = p / 2209;
    int r = p - b * 2209;
    int oy = r / 47;
    int ox = r - oy * 47;
    const float* inb = in + (size_t)b * 95 * 95 * 8;
    const unsigned char* mb = m + (size_t)b * 95 * 95;
    float acc[8];
#pragma unroll
    for (int c = 0; c < 8; ++c) acc[c] = -1e30f;
    bool any = false;
    for (int ky = 0; ky < 3; ++ky)
        for (int kx = 0; kx < 3; ++kx) {
            int q = (2 * oy + ky) * 95 + (2 * ox + kx);
            bool mk = mb[q] != 0;
            any |= mk;
            const float* v = inb + (size_t)q * 8;
#pragma unroll
            for (int c = 0; c < 8; ++c) {
                float x = v[c];
                acc[c] = fmaxf(acc[c], mk ? x : -1e30f);
            }
        }
    mo[p] = any ? 1 : 0;
    float* o = out + (size_t)p * 8;
#pragma unroll
    for (int c = 0; c < 8; ++c) o[c] = any ? acc[c] : 0.f;
}

// eval-mode BatchNorm + ReLU, masked (inactive -> 0). in may equal out.
__global__ void bnrelu_k(const float* __restrict__ in, float* __restrict__ out,
                         const unsigned char* __restrict__ m,
                         const float* __restrict__ sc, const float* __restrict__ bi,
                         const float* __restrict__ me, const float* __restrict__ va,
                         int P, int C) {
    int i = blockIdx.x * blockDim.x + threadIdx.x;
    if (i >= P * C) return;
    int c = i % C;
    int p = i / C;
    float a = sc[c] * (1.0f / sqrtf(va[c] + EPS_BN));
    float b = bi[c] - me[c] * a;
    float v = fmaf(in[i], a, b);
    v = v > 0.f ? v : 0.f;
    out[i] = m[p] ? v : 0.f;
}

// down_mask: any active site in FxF window, stride S, VALID
template <int F, int S>
__global__ void downmask_k(const unsigned char* __restrict__ m, unsigned char* __restrict__ mo,
                           int Hin, int Win, int Hout, int Wout) {
    int p = blockIdx.x * blockDim.x + threadIdx.x;
    int total = BB * Hout * Wout;
    if (p >= total) return;
    int hw = Hout * Wout;
    int b = p / hw;
    int r = p - b * hw;
    int oy = r / Wout;
    int ox = r - oy * Wout;
    const unsigned char* mb = m + (size_t)b * Hin * Win;
    bool any = false;
    for (int ky = 0; ky < F; ++ky)
        for (int kx = 0; kx < F; ++kx)
            any |= (mb[(oy * S + ky) * Win + (ox * S + kx)] != 0);
    mo[p] = any ? 1 : 0;
}

// =====================================================================
// Weight repack: HWIO w[(t*CINR+ci)*COUT+co] -> B-fragment order, padded
// over both input channels (ci>=CINR -> 0) and output channels (co>=COUT -> 0):
//   wT[pair*2*COUTP + 2*co + u] = w-value at kk = pair*2+u,  kk = t*CINP+ci.
// Result: per-lane B pair is one aligned guard-free b64 load in the GEMM.
// =====================================================================
__global__ void wtrans_k(const float* __restrict__ w, float* __restrict__ wT,
                         int KS2, int CINR, int CINP, int COUT, int COUTP) {
    int i = blockIdx.x * blockDim.x + threadIdx.x;
    int total = KS2 * CINP * COUTP;
    if (i >= total) return;
    int u = i & 1;
    int rest = i >> 1;
    int co = rest % COUTP;
    int pair = rest / COUTP;
    int kk = pair * 2 + u;
    int t = kk / CINP;
    int ci = kk - t * CINP;
    float v = 0.f;
    if (ci < CINR && co < COUT) v = w[(t * CINR + ci) * COUT + co];
    wT[i] = v;
}

// =====================================================================
// Implicit-GEMM WMMA conv (fp32, V_WMMA_F32_16X16X4_F32)
//   k = (ky*KS+kx)*CIN + ci  (HWIO, CIN possibly zero-padded).
//   One wave = 16(M pixels) x 16(N ch) tile. Branch-free inner loop:
//   clamped b64 A loads (+cndmask at SAME borders), padded b64 B loads.
//   Epilogue: optional BN+ReLU (bn_scale != null) XOR optional residual add.
// =====================================================================
template <int CIN, int COUT, int KS, int STRIDE, bool SAME>
__global__ void conv_wmma(const float* __restrict__ in, const float* __restrict__ wT,
                          float* __restrict__ out, const unsigned char* __restrict__ mask,
                          const float* __restrict__ bn_scale, const float* __restrict__ bn_bias,
                          const float* __restrict__ bn_mean, const float* __restrict__ bn_var,
                          const float* __restrict__ resid,
                          int Hin, int Win, int Hout, int Wout, int tilesN, int totalWaves) {
    static_assert((CIN % 4) == 0, "input channels must be padded to a multiple of 4");
    const int lane = threadIdx.x & 31;
    const int wid = blockIdx.x * (blockDim.x >> 5) + (threadIdx.x >> 5);
    if (wid >= totalWaves) return;
    const int tileM = wid / tilesN;
    const int tileN = wid - tileM * tilesN;

    constexpr int PAD = SAME ? (KS / 2) : 0;
    constexpr int COUTP = ((COUT + 15) / 16) * 16;
    constexpr bool FULLN = (COUT % 16) == 0;

    // A-fragment lane mapping: m = lane&15, k pair base = 2*(lane>>4)
    const int haln = lane >> 4;  // 0 or 1
    const int k0 = haln << 1;
    const int pA = tileM * 16 + (lane & 15);
    const int hw = Hout * Wout;
    const int bA = pA / hw;
    const int remA = pA - bA * hw;
    const int oy = remA / Wout;
    const int ox = remA - oy * Wout;
    const int iy0 = oy * STRIDE - PAD;
    const int ix0 = ox * STRIDE - PAD;
    const float* inB = in + (size_t)bA * Hin * Win * CIN;

    // B-fragment lane mapping
    const int co = tileN * 16 + (lane & 15);            // co < COUTP always
    const float* wTl = wT + 2 * co + haln * 2 * COUTP;  // + ks*4*COUTP per step

    v8f acc = {0.f, 0.f, 0.f, 0.f, 0.f, 0.f, 0.f, 0.f};

    // tap-major: hoist bounds/clamp per tap; per step one b64 A + one b64 B
    for (int t = 0; t < KS * KS; ++t) {
        const int dy = t / KS;
        const int dx = t - dy * KS;
        const int yy = iy0 + dy;
        const int xx = ix0 + dx;
        bool ok = true;
        int yyc = yy, xxc = xx;
        if constexpr (SAME) {
            ok = (yy >= 0) & (yy < Hin) & (xx >= 0) & (xx < Win);
            yyc = yy < 0 ? 0 : (yy >= Hin ? Hin - 1 : yy);
            xxc = xx < 0 ? 0 : (xx >= Win ? Win - 1 : xx);
        }
        const float* src = inB + (yyc * Win + xxc) * CIN + k0;
#pragma unroll
        for (int cb = 0; cb < CIN / 4; ++cb) {
            const int ks = t * (CIN / 4) + cb;
            v2f b = *(const v2f*)(wTl + (size_t)ks * 4 * COUTP);
            v2f raw = *(const v2f*)(src + cb * 4);
            v2f a;
            if constexpr (SAME) {
                a.x = ok ? raw.x : 0.f;
                a.y = ok ? raw.y : 0.f;
            } else {
                a = raw;
            }
            acc = __builtin_amdgcn_wmma_f32_16x16x4_f32(false, a, false, b, (short)0, acc,
                                                        false, false);
        }
    }

    // ---- epilogue (D row = v + 8*haln, col = lane&15)
    const bool coOK = FULLN || (co < COUT);
    const int coc = coOK ? co : 0;  // clamped for unconditional param loads
    const bool hasBN = (bn_scale != nullptr);
    float a_bn = 1.f, b_bn = 0.f;
    if (hasBN) {
        float inv = 1.0f / sqrtf(bn_var[coc] + EPS_BN);
        a_bn = bn_scale[coc] * inv;
        b_bn = bn_bias[coc] - bn_mean[coc] * a_bn;
    }
    const int rowHi = haln << 3;
#pragma unroll
    for (int v = 0; v < 8; ++v) {
        int p = tileM * 16 + (v + rowHi);
        float val = acc[v];
        if (hasBN) {
            val = fmaf(val, a_bn, b_bn);
            val = val > 0.f ? val : 0.f;
        }
        unsigned char mk = mask ? mask[p] : (unsigned char)1;
        size_t idx = (size_t)p * COUT + co;
        if constexpr (FULLN) {
            if (resid) val += resid[idx];
            out[idx] = mk ? val : 0.f;
        } else {
            if (coOK) {
                if (resid) val += resid[idx];
                out[idx] = mk ? val : 0.f;
            }
        }
    }
}

// =====================================================================
// Final linear: [512,64] @ [64,183] + bias (weights pre-repacked, COUTP=192)
// =====================================================================
__global__ void linear_wmma(const float* __restrict__ X, const float* __restrict__ WtT,
                            const float* __restrict__ bias, float* __restrict__ out,
                            int tilesN, int totalWaves) {
    const int lane = threadIdx.x & 31;
    const int wid = blockIdx.x * (blockDim.x >> 5) + (threadIdx.x >> 5);
    if (wid >= totalWaves) return;
    const int tileM = wid / tilesN;
    const int tileN = wid - tileM * tilesN;

    const int haln = lane >> 4;
    const int k0 = haln << 1;
    const int rowX = tileM * 16 + (lane & 15);
    const int co = tileN * 16 + (lane & 15);  // < 192
    const float* wTl = WtT + 2 * co + haln * 2 * 192;

    v8f acc = {0.f, 0.f, 0.f, 0.f, 0.f, 0.f, 0.f, 0.f};
#pragma unroll
    for (int ks = 0; ks < 16; ++ks) {  // K = 64
        v2f a = *(const v2f*)(X + rowX * 64 + ks * 4 + k0);
        v2f b = *(const v2f*)(wTl + (size_t)ks * 4 * 192);
        acc = __builtin_amdgcn_wmma_f32_16x16x4_f32(false, a, false, b, (short)0, acc,
                                                    false, false);
    }
    const bool coOK = (co < 183);
    const int coc = coOK ? co : 0;
    const float bco = bias[coc];
    const int rowHi = haln << 3;
#pragma unroll
    for (int v = 0; v < 8; ++v) {
        int p = tileM * 16 + (v + rowHi);
        if (coOK) out[(size_t)p * 183 + co] = acc[v] + bco;
    }
}

// =====================================================================
// Host-side orchestration
// =====================================================================
extern "C" void kernel_launch(void* const* d_in, const int* in_sizes, int n_in,
                              void* d_out, int out_size, void* d_ws, size_t ws_size,
                              hipStream_t stream) {
    if (n_in < 97) return;  // unexpected flattening; nothing sane to do

    const int* coords = (const int*)d_in[0];
    const float* feats = (const float*)d_in[1];
    const int NPTS = in_sizes[0] / 3;

    auto FP = [&](int i) -> const float* { return (const float*)d_in[i]; };
    // jax tree order (sorted dict keys at every level); BN dicts: bias,mean,scale,var
    const int BN_FINAL = 2, BN_OUT = 6;
    const int W_CONV5 = 10;
    const int G1R0_BNMID = 11, G1R0_BNPRE = 15, G1R0_W1 = 19, G1R0_W2 = 20;
    const int G1R1_BN1 = 21, G1R1_BN2 = 25, G1R1_W1 = 29, G1R1_W2 = 30;
    const int G2R0_BNMID = 31, G2R0_BNPRE = 35, G2R0_W1 = 39, G2R0_W2 = 40, G2R0_WRES = 41;
    const int G2R1_BN1 = 42, G2R1_BN2 = 46, G2R1_W1 = 50, G2R1_W2 = 51;
    const int G3R0_BNMID = 52, G3R0_BNPRE = 56, G3R0_W1 = 60, G3R0_W2 = 61, G3R0_WRES = 62;
    const int G3R1_BN1 = 63, G3R1_BN2 = 67, G3R1_W1 = 71, G3R1_W2 = 72;
    const int G4R0_BNMID = 73, G4R0_BNPRE = 77, G4R0_W1 = 81, G4R0_W2 = 82, G4R0_WRES = 83;
    const int G4R1_BN1 = 84, G4R1_BN2 = 88, G4R1_W1 = 92, G4R1_W2 = 93;
    const int LIN_B = 94, LIN_W = 95, W_SUBM0 = 96;

    // ---- workspace layout (bytes); heavy aliasing keeps total ~247 MB ----
    const size_t N0 = (size_t)BB * 95 * 95;  // 4,620,800
    const size_t P1 = (size_t)BB * 47 * 47;  // 1,131,008
    const size_t P2 = (size_t)BB * 23 * 23;  // 270,848
    const size_t P3 = (size_t)BB * 11 * 11;  // 61,952
    const size_t P4 = (size_t)BB * 5 * 5;    // 12,800

    char* ws = (char*)d_ws;
    const size_t OFF_FIRST = 0;                   // dead after build_x0
    const size_t OFF_M0 = OFF_FIRST + N0 * 4;
    const size_t OFF_X0 = OFF_M0 + N0;            // [512,95,95,4]; dead after subm0
    const size_t OFF_X1 = OFF_X0 + N0 * 4 * 4;    // [512,95,95,8]; dead after maxpool
    const size_t OFF_MASKS = OFF_X1 + N0 * 8 * 4;
    const size_t OFF_WT = OFF_MASKS + (P1 + P2 + P3 + P4 + 512 + 255) / 256 * 256;

    int* FIRST = (int*)(ws + OFF_FIRST);
    unsigned char* M0 = (unsigned char*)(ws + OFF_M0);
    float* X0 = (float*)(ws + OFF_X0);
    float* X1 = (float*)(ws + OFF_X1);
    unsigned char* M1 = (unsigned char*)(ws + OFF_MASKS);
    unsigned char* M2 = M1 + P1;
    unsigned char* M3 = M2 + P2;
    unsigned char* M4 = M3 + P3;
    unsigned char* M5 = M4 + P4;

    // level buffers (aliased into dead regions)
    float* L1A = (float*)(ws + OFF_X0);  // 36.2 MB <= X0 region (73.9 MB)
    float* L1B = (float*)(ws + OFF_X1 + 0);
    float* L1C = (float*)(ws + OFF_X1 + 36192256);
    float* L2A = (float*)(ws + OFF_X1 + 72384512);
    float* L2B = (float*)(ws + OFF_X1 + 89718784);
    float* L2C = (float*)(ws + OFF_X1 + 107053056);
    float* L4A = (float*)(ws + OFF_X1 + 124387328);
    float* L4B = (float*)(ws + OFF_X1 + 126025728);
    float* L4C = (float*)(ws + OFF_X1 + 127664128);
    float* X5 = (float*)(ws + OFF_X1 + 129302528);  // [512,64]
    float* L3A = (float*)(ws + OFF_FIRST);          // FIRST region dead by now
    float* L3B = (float*)(ws + OFF_FIRST + 5947392);
    float* L3C = (float*)(ws + OFF_FIRST + 11894784);

    // ---- weight repack allocator (tiny; runs every call, deterministic) ----
    float* wtBase = (float*)(ws + OFF_WT);
    size_t wtUsed = 0;
    auto WT = [&](int paramIdx, int KS2, int CINR, int CINP, int COUT) -> const float* {
        int COUTP = ((COUT + 15) / 16) * 16;
        int total = KS2 * CINP * COUTP;
        float* dst = wtBase + wtUsed;
        wtrans_k<<<(total + 255) / 256, 256, 0, stream>>>(FP(paramIdx), dst, KS2, CINR,
                                                          CINP, COUT, COUTP);
        wtUsed = (wtUsed + (size_t)total + 15) & ~(size_t)15;
        return dst;
    };

#define LAUNCH_CONV(CIR, CI, CO, KSZ, ST, SM, INP, WIDX, OUTP, MASKP, BNBASE, RES, HIN,     \
                    HOUT)                                                                   \
    do {                                                                                    \
        const float* wt_ = WT(WIDX, (KSZ) * (KSZ), CIR, CI, CO);                            \
        int P_ = BB * (HOUT) * (HOUT);                                                      \
        int tm_ = P_ / 16;                                                                  \
        int tn_ = ((CO) + 15) / 16;                                                         \
        int wv_ = tm_ * tn_;                                                                \
        int bl_ = (wv_ + 3) / 4;                                                            \
        conv_wmma<CI, CO, KSZ, ST, SM><<<bl_, 128, 0, stream>>>(                            \
            INP, wt_, OUTP, MASKP,                                                          \
            (BNBASE) >= 0 ? FP((BNBASE) + 2) : nullptr,                                     \
            (BNBASE) >= 0 ? FP((BNBASE) + 0) : nullptr,                                     \
            (BNBASE) >= 0 ? FP((BNBASE) + 1) : nullptr,                                     \
            (BNBASE) >= 0 ? FP((BNBASE) + 3) : nullptr,                                     \
            RES, HIN, HIN, HOUT, HOUT, tn_, wv_);                                           \
    } while (0)

    auto bnrelu = [&](const float* in_, float* out_, const unsigned char* m_, int base,
                      int P_, int C_) {
        int n = P_ * C_;
        bnrelu_k<<<(n + 255) / 256, 256, 0, stream>>>(in_, out_, m_, FP(base + 2),
                                                      FP(base + 0), FP(base + 1),
                                                      FP(base + 3), P_, C_);
    };

    // ---- 0) scatter points -> dense [512,95,95,4] + mask ----
    init_first_k<<<((int)N0 + 255) / 256, 256, 0, stream>>>(FIRST, (int)N0);
    scatter_k<<<(NPTS + 255) / 256, 256, 0, stream>>>(coords, FIRST, NPTS);
    build_x0_k<<<((int)N0 + 255) / 256, 256, 0, stream>>>(FIRST, feats, X0, M0, (int)N0);

    // ---- 1) subm0 conv 3(->4 padded)->8 @95, then maxpool -> 47 ----
    LAUNCH_CONV(3, 4, 8, 3, 1, true, X0, W_SUBM0, X1, M0, -1, nullptr, 95, 95);
    maxpool_k<<<((int)P1 + 255) / 256, 256, 0, stream>>>(X1, M0, L1A, M1);

    // ---- 2) group1 @47, 8ch ----
    bnrelu(L1A, L1A, M1, G1R0_BNPRE, (int)P1, 8);  // pre-act (shared w/ identity)
    LAUNCH_CONV(8, 8, 8, 3, 1, true, L1A, G1R0_W1, L1B, M1, G1R0_BNMID, nullptr, 47, 47);
    LAUNCH_CONV(8, 8, 8, 3, 1, true, L1B, G1R0_W2, L1A, M1, -1, L1A, 47, 47);
    bnrelu(L1A, L1B, M1, G1R1_BN1, (int)P1, 8);
    LAUNCH_CONV(8, 8, 8, 3, 1, true, L1B, G1R1_W1, L1C, M1, G1R1_BN2, nullptr, 47, 47);
    LAUNCH_CONV(8, 8, 8, 3, 1, true, L1C, G1R1_W2, L1A, M1, -1, L1A, 47, 47);

    // ---- 3) group2: stride-2 to 23, 16ch ----
    bnrelu(L1A, L1A, M1, G2R0_BNPRE, (int)P1, 8);
    downmask_k<3, 2><<<((int)P2 + 255) / 256, 256, 0, stream>>>(M1, M2, 47, 47, 23, 23);
    LAUNCH_CONV(8, 8, 16, 3, 2, false, L1A, G2R0_W1, L2A, M2, G2R0_BNMID, nullptr, 47, 23);
    LAUNCH_CONV(16, 16, 16, 3, 1, true, L2A, G2R0_W2, L2B, M2, -1, nullptr, 23, 23);
    LAUNCH_CONV(8, 8, 16, 3, 2, false, L1A, G2R0_WRES, L2B, M2, -1, L2B, 47, 23);
    bnrelu(L2B, L2A, M2, G2R1_BN1, (int)P2, 16);
    LAUNCH_CONV(16, 16, 16, 3, 1, true, L2A, G2R1_W1, L2C, M2, G2R1_BN2, nullptr, 23, 23);
    LAUNCH_CONV(16, 16, 16, 3, 1, true, L2C, G2R1_W2, L2B, M2, -1, L2B, 23, 23);

    // ---- 4) group3: stride-2 to 11, 24ch ----
    bnrelu(L2B, L2B, M2, G3R0_BNPRE, (int)P2, 16);
    downmask_k<3, 2><<<((int)P3 + 255) / 256, 256, 0, stream>>>(M2, M3, 23, 23, 11, 11);
    LAUNCH_CONV(16, 16, 24, 3, 2, false, L2B, G3R0_W1, L3A, M3, G3R0_BNMID, nullptr, 23, 11);
    LAUNCH_CONV(24, 24, 24, 3, 1, true, L3A, G3R0_W2, L3B, M3, -1, nullptr, 11, 11);
    LAUNCH_CONV(16, 16, 24, 3, 2, false, L2B, G3R0_WRES, L3B, M3, -1, L3B, 23, 11);
    bnrelu(L3B, L3A, M3, G3R1_BN1, (int)P3, 24);
    LAUNCH_CONV(24, 24, 24, 3, 1, true, L3A, G3R1_W1, L3C, M3, G3R1_BN2, nullptr, 11, 11);
    LAUNCH_CONV(24, 24, 24, 3, 1, true, L3C, G3R1_W2, L3B, M3, -1, L3B, 11, 11);

    // ---- 5) group4: stride-2 to 5, 32ch ----
    bnrelu(L3B, L3B, M3, G4R0_BNPRE, (int)P3, 24);
    downmask_k<3, 2><<<((int)P4 + 255) / 256, 256, 0, stream>>>(M3, M4, 11, 11, 5, 5);
    LAUNCH_CONV(24, 24, 32, 3, 2, false, L3B, G4R0_W1, L4A, M4, G4R0_BNMID, nullptr, 11, 5);
    LAUNCH_CONV(32, 32, 32, 3, 1, true, L4A, G4R0_W2, L4B, M4, -1, nullptr, 5, 5);
    LAUNCH_CONV(24, 24, 32, 3, 2, false, L3B, G4R0_WRES, L4B, M4, -1, L4B, 11, 5);
    bnrelu(L4B, L4A, M4, G4R1_BN1, (int)P4, 32);
    LAUNCH_CONV(32, 32, 32, 3, 1, true, L4A, G4R1_W1, L4C, M4, G4R1_BN2, nullptr, 5, 5);
    LAUNCH_CONV(32, 32, 32, 3, 1, true, L4C, G4R1_W2, L4B, M4, -1, L4B, 5, 5);

    // ---- 6) head: bn_final -> conv5(5x5,32->64,VALID)+bn_out -> linear ----
    bnrelu(L4B, L4B, M4, BN_FINAL, (int)P4, 32);
    downmask_k<5, 1><<<(BB + 255) / 256, 256, 0, stream>>>(M4, M5, 5, 5, 1, 1);
    LAUNCH_CONV(32, 32, 64, 5, 1, false, L4B, W_CONV5, X5, M5, BN_OUT, nullptr, 5, 1);

    {   // linear [512,64] x [64,183] + bias
        const float* wtl = WT(LIN_W, 1, 64, 64, 183);  // COUTP = 192
        int tn = (183 + 15) / 16;                      // 12
        int wv = (BB / 16) * tn;                       // 384
        int bl = (wv + 3) / 4;
        linear_wmma<<<bl, 128, 0, stream>>>(X5, wtl, FP(LIN_B), (float*)d_out, tn, wv);
    }
#undef LAUNCH_CONV
    (void)ws_size; (void)out_size;
}